// SpectConv_84112639525113
// MI455X (gfx1250) — compile-verified
//
#include <hip/hip_runtime.h>

#define N_NODES 10000
#define N_EDGES 320000
#define KFEAT   5
#define C_IN    128
#define C_OUT   128
#define KDIM    (KFEAT * C_IN)   // 640: reduction dim of the fused GEMM

typedef __attribute__((ext_vector_type(2))) float v2f;
typedef __attribute__((ext_vector_type(8))) float v8f;

// ---------------------------------------------------------------------------
// Kernel 0: zero the (N_NODES x KDIM) accumulator in workspace (f32).
// ---------------------------------------------------------------------------
__global__ void spect_zero(float4* __restrict__ p, int n4) {
    int i = blockIdx.x * blockDim.x + threadIdx.x;
    if (i < n4) {
        float4 z; z.x = 0.f; z.y = 0.f; z.z = 0.f; z.w = 0.f;
        p[i] = z;
    }
}

// ---------------------------------------------------------------------------
// Kernel 1: edge scatter.  One 128-thread block per edge (4 waves).
//   res[src][k*128 + c] += X[e][k] * h[dst][c]
// h row read is fully coalesced; X[e][0..4] and the int64 indices are staged
// through LDS once per edge.  Atomic adds are non-returning f32 -> resolved
// at L2 (res is 25.6 MB, L2-resident on the 192 MB L2).
// ---------------------------------------------------------------------------
__global__ void spect_scatter(const float* __restrict__ h,
                              const float* __restrict__ X,
                              const long long* __restrict__ edge_index,
                              float* __restrict__ res) {
    const int e = blockIdx.x;
    const int c = threadIdx.x;           // 0..127 channel

    __shared__ float xs[KFEAT];
    __shared__ int   s_src, s_dst;

    if (c == 0) {
        s_src = (int)edge_index[e];            // row 0: src
        s_dst = (int)edge_index[N_EDGES + e];  // row 1: dst
    }
    if (c < KFEAT) xs[c] = X[e * KFEAT + c];
    __syncthreads();

    const float hv = h[s_dst * C_IN + c];
    float* __restrict__ r = res + s_src * KDIM + c;
#pragma unroll
    for (int k = 0; k < KFEAT; ++k) {
        atomicAdd(r + k * C_OUT * 0 + k * C_IN, xs[k] * hv);
    }
}

// ---------------------------------------------------------------------------
// Kernel 2: out = res(10000x640) @ W(640x128) + bias, f32 WMMA.
// One wave per 16x16 output tile via V_WMMA_F32_16X16X4_F32 (K-step 4,
// 160 iterations).  256-thread block = 8 waves = one 16x128 output stripe.
//
// Fragment layouts (ISA 7.12.2, wave32):
//   A 16x4 f32 : lane holds row m=lane&15; VGPR v = K = kk + v + 2*(lane>>4)
//                -> contiguous float2 from row-major res.
//   B 4x16  f32: VGPR v = B[kk + v + 2*(lane>>4)][n], n = lane&15.
//   C/D 16x16  : VGPR v = out[m0 + v + 8*(lane>>4)][n].
// ---------------------------------------------------------------------------
__global__ void spect_gemm(const float* __restrict__ res,
                           const float* __restrict__ W,
                           const float* __restrict__ bias,
                           float* __restrict__ out) {
    const int wave = threadIdx.x >> 5;
    const int lane = threadIdx.x & 31;
    const int g    = lane >> 4;          // half-wave group (0 or 1)
    const int col  = lane & 15;

    const int m0 = blockIdx.x * 16;      // 625 blocks x 16 rows = 10000
    const int n  = wave * 16 + col;      // 8 waves x 16 cols = 128

    const float* __restrict__ arow = res + (m0 + col) * KDIM + 2 * g;

    v8f acc = {};
#pragma unroll 4
    for (int kk = 0; kk < KDIM; kk += 4) {
        v2f a = *(const v2f*)(arow + kk);                    // A[m][kk+2g], A[m][kk+2g+1]
        v2f b;
        b.x = W[(kk + 2 * g) * C_OUT + n];                   // B[kk+2g  ][n]
        b.y = W[(kk + 2 * g + 1) * C_OUT + n];               // B[kk+2g+1][n]
        // 8 args: (neg_a, A, neg_b, B, c_mod, C, reuse_a, reuse_b)
        acc = __builtin_amdgcn_wmma_f32_16x16x4_f32(
            false, a, false, b, (short)0, acc, false, false);
    }

    const float bv = bias[n];
#pragma unroll
    for (int v = 0; v < 8; ++v) {
        const int m = m0 + v + 8 * g;
        out[m * C_OUT + n] = acc[v] + bv;
    }
}

// ---------------------------------------------------------------------------
// Launch
// ---------------------------------------------------------------------------
extern "C" void kernel_launch(void* const* d_in, const int* in_sizes, int n_in,
                              void* d_out, int out_size, void* d_ws, size_t ws_size,
                              hipStream_t stream) {
    const float*     h    = (const float*)d_in[0];      // (10000,128)
    const float*     X    = (const float*)d_in[1];      // (320000,5)
    const long long* ei   = (const long long*)d_in[2];  // (2,320000) int64
    // d_in[3] = batch_node (unused by the reference math)
    const float*     W    = (const float*)d_in[4];      // (5,128,128) == (640,128)
    const float*     bias = (const float*)d_in[5];      // (128,)
    float*           out  = (float*)d_out;              // (10000,128)
    float*           res  = (float*)d_ws;               // (10000,640) scratch

    // 1) zero accumulator (25.6 MB in d_ws)
    {
        const int n4 = N_NODES * KDIM / 4;               // float4 count
        spect_zero<<<(n4 + 255) / 256, 256, 0, stream>>>((float4*)res, n4);
    }
    // 2) atomic scatter: one block per edge
    spect_scatter<<<N_EDGES, C_IN, 0, stream>>>(h, X, ei, res);
    // 3) fused einsum as f32 WMMA GEMM + bias
    spect_gemm<<<N_NODES / 16, 256, 0, stream>>>(res, W, bias, out);
}